// AdarEditGAT_50113678410338
// MI455X (gfx1250) — compile-verified
//
#include <hip/hip_runtime.h>
#include <hip/hip_bf16.h>

typedef __attribute__((ext_vector_type(2)))  float    v2f;
typedef __attribute__((ext_vector_type(8)))  float    v8f;

// ---------------------------------------------------------------- helpers
__device__ __forceinline__ float gelu_f(float x) {
  return 0.5f * x * (1.0f + erff(x * 0.70710678118654752440f));
}
__device__ __forceinline__ void atomicMaxFloat(float* addr, float val) {
  // valid when buffer initialized to -inf; signed/unsigned ordering trick
  if (val >= 0.0f) atomicMax((int*)addr, __float_as_int(val));
  else             atomicMin((unsigned int*)addr, __float_as_uint(val));
}
__device__ __forceinline__ void edge_sd(const int* __restrict__ ei, int E, int e,
                                        int& src, int& dst) {
  if (e < E) { src = ei[e]; dst = ei[E + e]; }
  else       { src = e - E; dst = e - E; }      // self loops appended
}

__global__ __launch_bounds__(256)
void fill_kernel(float* __restrict__ p, float v, long n) {
  long i = (long)blockIdx.x * 256 + threadIdx.x;
  if (i < n) p[i] = v;
}

// ------------------------------------------------- WMMA GEMM: Out = A @ W^T
// A: [Nn,K] row-major, W: [O,K] row-major, Out: [Nn,O].
// One wave computes a 16x64 strip: one A fragment feeds 4 independent WMMAs.
// No predicated loads: row index clamped (garbage rows never stored); K tail
// (K=21) handled by clamped load + value select on the B operand only.
template <int K>
__global__ __launch_bounds__(256)
void gemm_xWT_wmma(const float* __restrict__ A, const float* __restrict__ W,
                   float* __restrict__ Out, int Nn, int O) {
  const int lane = threadIdx.x & 31;
  const int wave = threadIdx.x >> 5;
  const int colGroups = O >> 6;                        // 64 output cols per wave
  const int rowTiles = (Nn + 15) >> 4;
  const long tile = (long)blockIdx.x * 8 + wave;       // uniform per wave
  if (tile >= (long)rowTiles * colGroups) return;      // whole-wave exit, EXEC all-1
  const int mt = (int)(tile / colGroups);
  const int ng = (int)(tile % colGroups);
  const int li = lane & 15;
  const int hi = lane >> 4;                            // lane half: 0 or 1
  int m = (mt << 4) + li;
  if (m >= Nn) m = Nn - 1;                             // clamp: loads valid, rows not stored
  const int nn0 = (ng << 6) + li;                      // columns nn0 + {0,16,32,48}

  const float* Arow = A + (long)m * K;
  const float* W0 = W + (long)nn0 * K;
  const float* W1 = W + (long)(nn0 + 16) * K;
  const float* W2 = W + (long)(nn0 + 32) * K;
  const float* W3 = W + (long)(nn0 + 48) * K;

  v8f acc0 = {}, acc1 = {}, acc2 = {}, acc3 = {};
  constexpr int Kfull = K & ~3;

  for (int kb = 0; kb < Kfull; kb += 4) {
    const int k0 = kb + (hi << 1);                     // lane half selects K pair
    v2f a, b0, b1, b2, b3;
    if constexpr ((K & 1) == 0) {
      // K even => (m*K + k0) even => 8B-aligned: emits global_load_b64
      a  = *(const v2f*)(Arow + k0);
      b0 = *(const v2f*)(W0 + k0);
      b1 = *(const v2f*)(W1 + k0);
      b2 = *(const v2f*)(W2 + k0);
      b3 = *(const v2f*)(W3 + k0);
    } else {
      a.x = Arow[k0]; a.y = Arow[k0 + 1];
      b0.x = W0[k0];  b0.y = W0[k0 + 1];
      b1.x = W1[k0];  b1.y = W1[k0 + 1];
      b2.x = W2[k0];  b2.y = W2[k0 + 1];
      b3.x = W3[k0];  b3.y = W3[k0 + 1];
    }
    acc0 = __builtin_amdgcn_wmma_f32_16x16x4_f32(false, a, false, b0, (short)0, acc0, false, false);
    acc1 = __builtin_amdgcn_wmma_f32_16x16x4_f32(false, a, false, b1, (short)0, acc1, false, false);
    acc2 = __builtin_amdgcn_wmma_f32_16x16x4_f32(false, a, false, b2, (short)0, acc2, false, false);
    acc3 = __builtin_amdgcn_wmma_f32_16x16x4_f32(false, a, false, b3, (short)0, acc3, false, false);
  }
  if constexpr (Kfull < K) {
    // tail: unconditional clamped loads; zero B via v_cndmask (A garbage * 0 = 0)
    const int k0 = Kfull + (hi << 1);
    const int k1 = k0 + 1;
    const int k0c = (k0 < K) ? k0 : (K - 1);
    const int k1c = (k1 < K) ? k1 : (K - 1);
    v2f a, b0, b1, b2, b3;
    a.x = Arow[k0c]; a.y = Arow[k1c];
    float w;
    w = W0[k0c]; b0.x = (k0 < K) ? w : 0.0f;
    w = W0[k1c]; b0.y = (k1 < K) ? w : 0.0f;
    w = W1[k0c]; b1.x = (k0 < K) ? w : 0.0f;
    w = W1[k1c]; b1.y = (k1 < K) ? w : 0.0f;
    w = W2[k0c]; b2.x = (k0 < K) ? w : 0.0f;
    w = W2[k1c]; b2.y = (k1 < K) ? w : 0.0f;
    w = W3[k0c]; b3.x = (k0 < K) ? w : 0.0f;
    w = W3[k1c]; b3.y = (k1 < K) ? w : 0.0f;
    acc0 = __builtin_amdgcn_wmma_f32_16x16x4_f32(false, a, false, b0, (short)0, acc0, false, false);
    acc1 = __builtin_amdgcn_wmma_f32_16x16x4_f32(false, a, false, b1, (short)0, acc1, false, false);
    acc2 = __builtin_amdgcn_wmma_f32_16x16x4_f32(false, a, false, b2, (short)0, acc2, false, false);
    acc3 = __builtin_amdgcn_wmma_f32_16x16x4_f32(false, a, false, b3, (short)0, acc3, false, false);
  }

  // D layout: lane holds column (nn), VGPR v -> row v + 8*hi
  const int mb = (mt << 4) + (hi << 3);
  for (int v = 0; v < 8; ++v) {
    const int mr = mb + v;
    if (mr < Nn) {
      long ro = (long)mr * O;
      Out[ro + nn0]      = acc0[v];
      Out[ro + nn0 + 16] = acc1[v];
      Out[ro + nn0 + 32] = acc2[v];
      Out[ro + nn0 + 48] = acc3[v];
    }
  }
}

// ------------------------------------------------- GAT attention pipeline
__global__ __launch_bounds__(256)
void sd_kernel(const float* __restrict__ hlin, const float* __restrict__ as_,
               const float* __restrict__ ad_, float* __restrict__ s,
               float* __restrict__ d, int Nn) {
  int tid = blockIdx.x * 256 + threadIdx.x;
  if (tid >= Nn * 4) return;
  int n = tid >> 2, h = tid & 3;
  const float* hp = hlin + (long)n * 256 + h * 64;
  const float* ap = as_ + h * 64;
  const float* dp = ad_ + h * 64;
  float ss = 0.0f, dd = 0.0f;
  for (int c = 0; c < 64; ++c) { float v = hp[c]; ss += v * ap[c]; dd += v * dp[c]; }
  s[tid] = ss; d[tid] = dd;
}

__global__ __launch_bounds__(256)
void edge_logits_kernel(const int* __restrict__ ei, int E, int Ep,
                        const float* __restrict__ s, const float* __restrict__ d,
                        float* __restrict__ ebuf, float* __restrict__ m) {
  int tid = blockIdx.x * 256 + threadIdx.x;
  if (tid >= Ep * 4) return;
  int e = tid >> 2, h = tid & 3, src, dst;
  edge_sd(ei, E, e, src, dst);
  float v = s[src * 4 + h] + d[dst * 4 + h];
  v = (v > 0.0f) ? v : 0.2f * v;                        // leaky_relu 0.2
  ebuf[tid] = v;
  atomicMaxFloat(&m[dst * 4 + h], v);
}

__global__ __launch_bounds__(256)
void edge_exp_kernel(const int* __restrict__ ei, int E, int Ep,
                     const float* __restrict__ m, float* __restrict__ ebuf,
                     float* __restrict__ z) {
  int tid = blockIdx.x * 256 + threadIdx.x;
  if (tid >= Ep * 4) return;
  int e = tid >> 2, h = tid & 3, src, dst;
  edge_sd(ei, E, e, src, dst);
  float ex = expf(ebuf[tid] - m[dst * 4 + h]);
  ebuf[tid] = ex;
  atomicAdd(&z[dst * 4 + h], ex);
}

__global__ __launch_bounds__(256)
void edge_scatter_kernel(const int* __restrict__ ei, int E, int Ep,
                         const float* __restrict__ hlin, const float* __restrict__ ebuf,
                         const float* __restrict__ z, float* __restrict__ acc) {
  long tid = (long)blockIdx.x * 256 + threadIdx.x;
  if (tid >= (long)Ep * 16) return;
  int e = (int)(tid >> 4), q = (int)(tid & 15);        // q = 4-float chunk of 64
  int src, dst;
  edge_sd(ei, E, e, src, dst);
  const int c4 = q * 4;
  for (int h = 0; h < 4; ++h) {
    float alpha = ebuf[e * 4 + h] / z[dst * 4 + h];
    const float4 hv = *(const float4*)(hlin + (long)src * 256 + h * 64 + c4);
    float* ap = acc + (long)dst * 256 + h * 64 + c4;
    atomicAdd(ap + 0, hv.x * alpha);
    atomicAdd(ap + 1, hv.y * alpha);
    atomicAdd(ap + 2, hv.z * alpha);
    atomicAdd(ap + 3, hv.w * alpha);
  }
}

__global__ __launch_bounds__(256)
void finalize_kernel(const float* __restrict__ acc, const float* __restrict__ bias,
                     float* __restrict__ feat, int Nn) {
  int tid = blockIdx.x * 256 + threadIdx.x;
  if (tid >= Nn * 64) return;
  int n = tid >> 6, c = tid & 63;
  const float* p = acc + (long)n * 256 + c;
  float v = (p[0] + p[64] + p[128] + p[192]) * 0.25f + bias[c];  // head mean + bias
  feat[tid] = gelu_f(v);
}

// ------------------------------------------------- pooling (softmax over ALL nodes)
__global__ __launch_bounds__(256)
void pool_logit_kernel(const float* __restrict__ feat, const float* __restrict__ pw,
                       const float* __restrict__ pb, float* __restrict__ t, int Nn) {
  int n = blockIdx.x * 256 + threadIdx.x;
  if (n >= Nn) return;
  float a = pb[0];
  for (int c = 0; c < 64; ++c) a += feat[(long)n * 64 + c] * pw[c];
  t[n] = a;
}

__global__ __launch_bounds__(256)
void reduce_max_kernel(const float* __restrict__ t, int n, float* __restrict__ out) {
  __shared__ float sm[256];
  float v = -INFINITY;
  for (int i = blockIdx.x * 256 + threadIdx.x; i < n; i += gridDim.x * 256)
    v = fmaxf(v, t[i]);
  sm[threadIdx.x] = v; __syncthreads();
  for (int s = 128; s > 0; s >>= 1) {
    if (threadIdx.x < s) sm[threadIdx.x] = fmaxf(sm[threadIdx.x], sm[threadIdx.x + s]);
    __syncthreads();
  }
  if (threadIdx.x == 0) atomicMaxFloat(out, sm[0]);
}

__global__ __launch_bounds__(256)
void reduce_sumexp_kernel(const float* __restrict__ t, int n,
                          const float* __restrict__ mx, float* __restrict__ out) {
  __shared__ float sm[256];
  float mv = mx[0];
  float v = 0.0f;
  for (int i = blockIdx.x * 256 + threadIdx.x; i < n; i += gridDim.x * 256)
    v += expf(t[i] - mv);
  sm[threadIdx.x] = v; __syncthreads();
  for (int s = 128; s > 0; s >>= 1) {
    if (threadIdx.x < s) sm[threadIdx.x] += sm[threadIdx.x + s];
    __syncthreads();
  }
  if (threadIdx.x == 0) atomicAdd(out, sm[0]);
}

__global__ __launch_bounds__(256)
void pool_scatter_kernel(const float* __restrict__ feat, const float* __restrict__ t,
                         const float* __restrict__ sc, const int* __restrict__ batch,
                         float* __restrict__ pooled, int Nn) {
  int tid = blockIdx.x * 256 + threadIdx.x;
  if (tid >= Nn * 64) return;
  int n = tid >> 6, c = tid & 63;
  float w = expf(t[n] - sc[0]) / sc[1];
  atomicAdd(&pooled[batch[n] * 64 + c], feat[tid] * w);
}

// ------------------------------------------------- MLP heads (B = 800)
__global__ __launch_bounds__(256)
void fused_e1_kernel(const float* __restrict__ pooled, const float* __restrict__ rnafm,
                     const float* __restrict__ hand, const float* __restrict__ W,
                     const float* __restrict__ b, float* __restrict__ z1, int B) {
  int tid = blockIdx.x * 256 + threadIdx.x;
  if (tid >= B * 256) return;
  int bb = tid >> 8, o = tid & 255;
  const float* wr = W + (long)o * 744;
  float a = b[o];
  for (int k = 0; k < 64; ++k)  a += pooled[bb * 64 + k]  * wr[k];
  for (int k = 0; k < 640; ++k) a += rnafm[bb * 640 + k]  * wr[64 + k];
  for (int k = 0; k < 40; ++k)  a += hand[bb * 40 + k]    * wr[704 + k];
  z1[tid] = gelu_f(a);
}

__global__ __launch_bounds__(256)
void layernorm_kernel(float* __restrict__ z1, const float* __restrict__ g,
                      const float* __restrict__ bta) {
  __shared__ float s1[256], s2[256];
  int bb = blockIdx.x, o = threadIdx.x;
  float v = z1[bb * 256 + o];
  s1[o] = v; s2[o] = v * v; __syncthreads();
  for (int st = 128; st > 0; st >>= 1) {
    if (o < st) { s1[o] += s1[o + st]; s2[o] += s2[o + st]; }
    __syncthreads();
  }
  float mu = s1[0] * (1.0f / 256.0f);
  float var = s2[0] * (1.0f / 256.0f) - mu * mu;
  float r = rsqrtf(var + 1e-5f);
  z1[bb * 256 + o] = (v - mu) * r * g[o] + bta[o];
}

__global__ __launch_bounds__(256)
void e2_kernel(const float* __restrict__ z1, const float* __restrict__ W,
               const float* __restrict__ b, float* __restrict__ sh, int B) {
  int tid = blockIdx.x * 256 + threadIdx.x;
  if (tid >= B * 128) return;
  int bb = tid >> 7, o = tid & 127;
  const float* wr = W + (long)o * 256;
  const float* zr = z1 + (long)bb * 256;
  float a = b[o];
  for (int k = 0; k < 256; ++k) a += zr[k] * wr[k];
  sh[tid] = gelu_f(a);
}

__global__ __launch_bounds__(256)
void bin_kernel(const float* __restrict__ sh, const float* __restrict__ W,
                const float* __restrict__ b, float* __restrict__ out, int B) {
  int bb = blockIdx.x * 256 + threadIdx.x;
  if (bb >= B) return;
  float a = b[0];
  for (int k = 0; k < 128; ++k) a += sh[bb * 128 + k] * W[k];
  out[bb] = a;
}

__global__ __launch_bounds__(256)
void ad1_kernel(const float* __restrict__ sh, const float* __restrict__ W,
                const float* __restrict__ b, float* __restrict__ h1, int B) {
  int tid = blockIdx.x * 256 + threadIdx.x;
  if (tid >= 5 * B * 32) return;
  int e = tid / (B * 32);
  int r = tid - e * (B * 32);
  int bb = r >> 5, hh = r & 31;
  const float* wr = W + ((long)e * 32 + hh) * 128;
  float a = b[e * 32 + hh];
  for (int k = 0; k < 128; ++k) a += sh[bb * 128 + k] * wr[k];
  h1[tid] = gelu_f(a);
}

__global__ __launch_bounds__(256)
void ad2_kernel(const float* __restrict__ h1, const float* __restrict__ W,
                const float* __restrict__ b, float* __restrict__ out, int B) {
  int tid = blockIdx.x * 256 + threadIdx.x;
  if (tid >= 5 * B) return;
  int e = tid / B, bb = tid - e * B;
  float a = b[e];
  const float* hr = h1 + ((long)e * B + bb) * 32;
  for (int k = 0; k < 32; ++k) a += hr[k] * W[e * 32 + k];
  out[e * B + bb] = a;          // per[e,b], caller offsets by B
}

__global__ __launch_bounds__(256)
void c1_kernel(const float* __restrict__ sh, const float* __restrict__ W,
               const float* __restrict__ b, float* __restrict__ ct, int B) {
  int tid = blockIdx.x * 256 + threadIdx.x;
  if (tid >= B * 64) return;
  int bb = tid >> 6, o = tid & 63;
  const float* wr = W + (long)o * 128;
  float a = b[o];
  for (int k = 0; k < 128; ++k) a += sh[bb * 128 + k] * wr[k];
  ct[tid] = gelu_f(a);
}

__global__ __launch_bounds__(256)
void c2_kernel(const float* __restrict__ ct, const float* __restrict__ W,
               const float* __restrict__ b, float* __restrict__ out, int B) {
  int tid = blockIdx.x * 256 + threadIdx.x;
  if (tid >= B * 6) return;
  int bb = tid / 6, j = tid - bb * 6;
  float a = b[j];
  for (int k = 0; k < 64; ++k) a += ct[bb * 64 + k] * W[j * 64 + k];
  out[bb * 6 + j] = a;          // cls[b,j], caller offsets by 6B
}

// ---------------------------------------------------------------- host
static inline dim3 hgrid(long n) { return dim3((unsigned)((n + 255) / 256)); }

extern "C" void kernel_launch(void* const* d_in, const int* in_sizes, int n_in,
                              void* d_out, int out_size, void* d_ws, size_t ws_size,
                              hipStream_t stream) {
  (void)n_in; (void)out_size; (void)ws_size;
  const int N  = in_sizes[0] / 21;
  const int E  = in_sizes[1] / 2;
  const int B  = in_sizes[3] / 640;
  const int Ep = E + N;

  const float* x     = (const float*)d_in[0];
  const int*   ei    = (const int*)d_in[1];
  const int*   batch = (const int*)d_in[2];
  const float* rnafm = (const float*)d_in[3];
  const float* hand  = (const float*)d_in[4];
  const float* gW[3]  = {(const float*)d_in[5],  (const float*)d_in[9],  (const float*)d_in[13]};
  const float* gAS[3] = {(const float*)d_in[6],  (const float*)d_in[10], (const float*)d_in[14]};
  const float* gAD[3] = {(const float*)d_in[7],  (const float*)d_in[11], (const float*)d_in[15]};
  const float* gB[3]  = {(const float*)d_in[8],  (const float*)d_in[12], (const float*)d_in[16]};
  const float* poolW = (const float*)d_in[17]; const float* poolB = (const float*)d_in[18];
  const float* e1W = (const float*)d_in[19];   const float* e1b = (const float*)d_in[20];
  const float* lnG = (const float*)d_in[21];   const float* lnB = (const float*)d_in[22];
  const float* e2W = (const float*)d_in[23];   const float* e2b = (const float*)d_in[24];
  const float* binW = (const float*)d_in[25];  const float* binB = (const float*)d_in[26];
  const float* ad1W = (const float*)d_in[27];  const float* ad1b = (const float*)d_in[28];
  const float* ad2W = (const float*)d_in[29];  const float* ad2b = (const float*)d_in[30];
  const float* c1W = (const float*)d_in[31];   const float* c1b = (const float*)d_in[32];
  const float* c2W = (const float*)d_in[33];   const float* c2b = (const float*)d_in[34];

  float* ws = (float*)d_ws;
  size_t off = 0;
  float* hlin   = ws + off; off += (size_t)N * 256;
  float* acc    = ws + off; off += (size_t)N * 256;
  float* feat   = ws + off; off += (size_t)N * 64;
  float* sbuf   = ws + off; off += (size_t)N * 4;
  float* dbuf   = ws + off; off += (size_t)N * 4;
  float* mbuf   = ws + off; off += (size_t)N * 4;
  float* zbuf   = ws + off; off += (size_t)N * 4;
  float* ebuf   = ws + off; off += (size_t)Ep * 4;
  float* tbuf   = ws + off; off += (size_t)N;
  float* pooled = ws + off; off += (size_t)B * 64;
  float* z1     = ws + off; off += (size_t)B * 256;
  float* shd    = ws + off; off += (size_t)B * 128;
  float* h1     = ws + off; off += (size_t)5 * B * 32;
  float* ctmp   = ws + off; off += (size_t)B * 64;
  float* sc     = ws + off; off += 2;

  float* out = (float*)d_out;

  const int strips = ((N + 15) / 16) * (256 / 64);     // row tiles * col groups
  const dim3 ggrid((unsigned)((strips + 7) / 8));

  for (int L = 0; L < 3; ++L) {
    if (L == 0)
      gemm_xWT_wmma<21><<<ggrid, dim3(256), 0, stream>>>(x,    gW[0], hlin, N, 256);
    else
      gemm_xWT_wmma<64><<<ggrid, dim3(256), 0, stream>>>(feat, gW[L], hlin, N, 256);
    sd_kernel<<<hgrid((long)N * 4), dim3(256), 0, stream>>>(hlin, gAS[L], gAD[L], sbuf, dbuf, N);
    fill_kernel<<<hgrid((long)N * 256), dim3(256), 0, stream>>>(acc, 0.0f, (long)N * 256);
    fill_kernel<<<hgrid((long)N * 4), dim3(256), 0, stream>>>(mbuf, -INFINITY, (long)N * 4);
    fill_kernel<<<hgrid((long)N * 4), dim3(256), 0, stream>>>(zbuf, 0.0f, (long)N * 4);
    edge_logits_kernel<<<hgrid((long)Ep * 4), dim3(256), 0, stream>>>(ei, E, Ep, sbuf, dbuf, ebuf, mbuf);
    edge_exp_kernel<<<hgrid((long)Ep * 4), dim3(256), 0, stream>>>(ei, E, Ep, mbuf, ebuf, zbuf);
    edge_scatter_kernel<<<hgrid((long)Ep * 16), dim3(256), 0, stream>>>(ei, E, Ep, hlin, ebuf, zbuf, acc);
    finalize_kernel<<<hgrid((long)N * 64), dim3(256), 0, stream>>>(acc, gB[L], feat, N);
  }

  pool_logit_kernel<<<hgrid(N), dim3(256), 0, stream>>>(feat, poolW, poolB, tbuf, N);
  fill_kernel<<<dim3(1), dim3(256), 0, stream>>>(sc, -INFINITY, 1);
  fill_kernel<<<dim3(1), dim3(256), 0, stream>>>(sc + 1, 0.0f, 1);
  reduce_max_kernel<<<dim3(256), dim3(256), 0, stream>>>(tbuf, N, sc);
  reduce_sumexp_kernel<<<dim3(256), dim3(256), 0, stream>>>(tbuf, N, sc, sc + 1);
  fill_kernel<<<hgrid((long)B * 64), dim3(256), 0, stream>>>(pooled, 0.0f, (long)B * 64);
  pool_scatter_kernel<<<hgrid((long)N * 64), dim3(256), 0, stream>>>(feat, tbuf, sc, batch, pooled, N);

  fused_e1_kernel<<<hgrid((long)B * 256), dim3(256), 0, stream>>>(pooled, rnafm, hand, e1W, e1b, z1, B);
  layernorm_kernel<<<dim3((unsigned)B), dim3(256), 0, stream>>>(z1, lnG, lnB);
  e2_kernel<<<hgrid((long)B * 128), dim3(256), 0, stream>>>(z1, e2W, e2b, shd, B);
  bin_kernel<<<hgrid(B), dim3(256), 0, stream>>>(shd, binW, binB, out, B);
  ad1_kernel<<<hgrid((long)5 * B * 32), dim3(256), 0, stream>>>(shd, ad1W, ad1b, h1, B);
  ad2_kernel<<<hgrid((long)5 * B), dim3(256), 0, stream>>>(h1, ad2W, ad2b, out + B, B);
  c1_kernel<<<hgrid((long)B * 64), dim3(256), 0, stream>>>(shd, c1W, c1b, ctmp, B);
  c2_kernel<<<hgrid((long)B * 6), dim3(256), 0, stream>>>(ctmp, c2W, c2b, out + 6 * B, B);
}